// ImprovedVectorQuantizer_19396072309141
// MI455X (gfx1250) — compile-verified
//
#include <hip/hip_runtime.h>

typedef __attribute__((ext_vector_type(2))) float v2f;
typedef __attribute__((ext_vector_type(8))) float v8f;

#define B_  64
#define D_  64
#define T_  2048
#define K_  512
#define N_  (B_ * T_)          // 131072
#define QELEMS (B_ * D_ * T_)  // 8388608

// ---------------------------------------------------------------------------
// Kernel 0: init — per-codebook-row squared norms, zero counters/accumulator
// ---------------------------------------------------------------------------
__global__ __launch_bounds__(512) void vq_init(const float* __restrict__ W,
                                               float* __restrict__ enorm,
                                               int* __restrict__ counts,
                                               float* __restrict__ lossAcc) {
    int k = threadIdx.x;  // 512 threads, one codebook row each
    float s = 0.0f;
    const float* wr = W + k * D_;
    #pragma unroll 8
    for (int d = 0; d < D_; ++d) { float w = wr[d]; s += w * w; }
    enorm[k]  = s;
    counts[k] = 0;
    if (k == 0) *lossAcc = 0.0f;
}

// ---------------------------------------------------------------------------
// Kernel 1: WMMA distance + argmin + FUSED quantize/commitment-loss.
// Each wave handles 16 rows (16 consecutive t within one b). 32 column tiles
// of 16 codebook entries; 16 chained V_WMMA_F32_16X16X4_F32 per tile (K=64).
// Codebook staged once per block in LDS with row stride 68 floats ->
// conflict-free ds_load_b64 fragment reads. After the argmin butterfly the
// winning index is broadcast back to the row-owning lanes, which quantize
// from LDS against the in-register A fragments — X is read exactly once.
// ---------------------------------------------------------------------------
__global__ __launch_bounds__(256) void vq_argmin(const float* __restrict__ X,
                                                 const float* __restrict__ W,
                                                 const float* __restrict__ enorm,
                                                 int* __restrict__ idxOut,
                                                 int* __restrict__ counts,
                                                 float* __restrict__ qOut,
                                                 float* __restrict__ lossAcc) {
    __shared__ float wlds[K_ * 68];  // 139264 B of the 320 KB WGP LDS

    // Cooperative, coalesced load of the whole codebook into padded LDS.
    for (int e = threadIdx.x; e < K_ * D_; e += 256) {
        int r = e >> 6, d = e & 63;
        wlds[r * 68 + d] = W[e];
    }
    __syncthreads();

    const int wave = threadIdx.x >> 5;   // 0..7
    const int lane = threadIdx.x & 31;
    const int half = lane >> 4;          // 0 | 1
    const int lr   = lane & 15;
    const int row0 = (blockIdx.x * 8 + wave) * 16;  // first flat row n of tile

    // A fragments: lane (lr,half) owns row M=lr; half 0 -> K=4k,4k+1 ; half 1 -> K=4k+2,4k+3
    const int n = row0 + lr;
    const int b = n >> 11;               // n / T
    const int t = n & (T_ - 1);
    const float* xrow = X + (size_t)b * (D_ * T_) + t;   // element d at xrow[d*T_]
    const int dbase = half * 2;

    v2f a[16];
    #pragma unroll
    for (int k = 0; k < 16; ++k) {
        int d0 = 4 * k + dbase;
        a[k].x = xrow[d0 * T_];
        a[k].y = xrow[(d0 + 1) * T_];
    }

    float bestVal[8];
    int   bestIdx[8];
    #pragma unroll
    for (int v = 0; v < 8; ++v) { bestVal[v] = 3.4e38f; bestIdx[v] = 0x7fffffff; }

    for (int c = 0; c < 32; ++c) {
        const int col = c * 16 + lr;                 // this lane's column (all 8 acc rows)
        const float* wrow = &wlds[col * 68 + dbase];
        v8f acc = {};
        #pragma unroll
        for (int k = 0; k < 16; ++k) {
            v2f bb;
            bb.x = wrow[4 * k];
            bb.y = wrow[4 * k + 1];
            acc = __builtin_amdgcn_wmma_f32_16x16x4_f32(
                false, a[k], false, bb, (short)0, acc, false, false);
        }
        const float en = enorm[col];
        #pragma unroll
        for (int v = 0; v < 8; ++v) {
            float dist = en - 2.0f * acc[v];   // ||x||^2 dropped: row-constant
            if (dist < bestVal[v] || (dist == bestVal[v] && col < bestIdx[v])) {
                bestVal[v] = dist;
                bestIdx[v] = col;
            }
        }
    }

    // Cross-lane argmin within each 16-lane half (C layout: N = lane&15),
    // then route each row's winning index to the lane that owns that row.
    int myIdx = 0;
    #pragma unroll
    for (int v = 0; v < 8; ++v) {
        float val = bestVal[v];
        int   bi  = bestIdx[v];
        #pragma unroll
        for (int off = 8; off >= 1; off >>= 1) {
            float oval = __shfl_xor(val, off, 32);
            int   oidx = __shfl_xor(bi,  off, 32);
            if (oval < val || (oval == val && oidx < bi)) { val = oval; bi = oidx; }
        }
        if (lr == 0) {                       // one writer per half per row
            int row = row0 + v + 8 * half;   // C row M = v + 8*(lane>=16)
            idxOut[row] = bi;
            atomicAdd(&counts[bi], 1);
        }
        int rLow  = __shfl(bi, 0,  32);      // reduced index of row v   (half 0)
        int rHigh = __shfl(bi, 16, 32);      // reduced index of row v+8 (half 1)
        if (lr == v)     myIdx = rLow;       // lane owns row lr (both halves)
        if (lr == v + 8) myIdx = rHigh;
    }

    // Fused quantize + commitment loss: codebook row from LDS, x from the
    // in-register A fragments, NT-store quantized back in [B, D, T] layout.
    const float* wq   = &wlds[myIdx * 68 + dbase];
    float*       qrow = qOut + (size_t)b * (D_ * T_) + t;
    float lsum = 0.0f;
    #pragma unroll
    for (int k = 0; k < 16; ++k) {
        int d0 = 4 * k + dbase;
        float q0 = wq[4 * k];
        float q1 = wq[4 * k + 1];
        float e0 = q0 - a[k].x;
        float e1 = q1 - a[k].y;
        lsum += e0 * e0 + e1 * e1;
        __builtin_nontemporal_store(q0, &qrow[d0 * T_]);
        __builtin_nontemporal_store(q1, &qrow[(d0 + 1) * T_]);
    }
    #pragma unroll
    for (int off = 16; off >= 1; off >>= 1) lsum += __shfl_xor(lsum, off, 32);
    if (lane == 0) atomicAdd(lossAcc, lsum);
}

// ---------------------------------------------------------------------------
// Kernel 2: one-hot encodings, 8 floats per thread, branchless, NT b64 stores
// (output base is only 8-byte aligned: offset 8388610 floats).
// ---------------------------------------------------------------------------
__global__ __launch_bounds__(256) void vq_enc(const int* __restrict__ idx,
                                              float* __restrict__ encOut) {
    unsigned tid = blockIdx.x * 256u + threadIdx.x;  // < N*64 = 8388608
    int chunk = tid & 63;                            // 8-float chunk within row
    int n     = tid >> 6;
    int e     = idx[n];
    int kb    = chunk << 3;
    float* base = encOut + ((size_t)n << 9) + kb;
    #pragma unroll
    for (int j = 0; j < 4; ++j) {
        v2f p;
        p.x = (e == kb + 2 * j)     ? 1.0f : 0.0f;
        p.y = (e == kb + 2 * j + 1) ? 1.0f : 0.0f;
        __builtin_nontemporal_store(p, (v2f*)(base + 2 * j));
    }
}

// ---------------------------------------------------------------------------
// Kernel 3: finalize scalars — loss and perplexity.
// ---------------------------------------------------------------------------
__global__ __launch_bounds__(512) void vq_final(const int* __restrict__ counts,
                                                const float* __restrict__ lossAcc,
                                                float* __restrict__ outLoss,
                                                float* __restrict__ outPerp) {
    __shared__ float red[512];
    int k = threadIdx.x;
    float p = (float)counts[k] * (1.0f / (float)N_);
    red[k] = -p * __logf(p + 1e-10f);
    __syncthreads();
    #pragma unroll
    for (int s = 256; s >= 1; s >>= 1) {
        if (k < s) red[k] += red[k + s];
        __syncthreads();
    }
    if (k == 0) {
        *outPerp = __expf(red[0]);
        *outLoss = 0.25f * (*lossAcc) * (1.0f / (float)QELEMS);
    }
}

// ---------------------------------------------------------------------------
extern "C" void kernel_launch(void* const* d_in, const int* in_sizes, int n_in,
                              void* d_out, int out_size, void* d_ws, size_t ws_size,
                              hipStream_t stream) {
    const float* X = (const float*)d_in[0];   // [B, D, T] fp32
    const float* W = (const float*)d_in[1];   // [K, D]   fp32
    float* out = (float*)d_out;

    char* ws = (char*)d_ws;
    int*   idxBuf  = (int*)ws;                  // N ints      (524288 B)
    int*   counts  = (int*)(ws + 524288);       // 512 ints
    float* enorm   = (float*)(ws + 526336);     // 512 floats
    float* lossAcc = (float*)(ws + 528384);     // 1 float

    float* lossOut = out;                 // [0]
    float* qOut    = out + 1;             // [1 .. 1+8388608)
    float* perpOut = out + 1 + QELEMS;    // [8388609]
    float* encOut  = out + 2 + QELEMS;    // [8388610 ..), 8B-aligned

    vq_init  <<<1,            512, 0, stream>>>(W, enorm, counts, lossAcc);
    vq_argmin<<<N_ / 128,     256, 0, stream>>>(X, W, enorm, idxBuf, counts,
                                                qOut, lossAcc);
    vq_enc   <<<QELEMS / 256, 256, 0, stream>>>(idxBuf, encOut);
    vq_final <<<1,            512, 0, stream>>>(counts, lossAcc, lossOut, perpOut);
}